// RoIPooling_9594956939847
// MI455X (gfx1250) — compile-verified
//
#include <hip/hip_runtime.h>

#define NCH   256
#define FM    7
#define OUTP  7
#define SCALE 32
#define FEAT        (NCH * FM * FM)      // 12544 floats = 49 KB
#define OUT_PER_ROI (NCH * OUTP * OUTP)  // 12544 floats per ROI (= 49 * 256)
#define TPB   256

// One block per ROI. Stage the feature map into LDS with CDNA5 async-to-LDS
// DMA (ASYNCcnt + s_wait_asynccnt), then branchless 2x2 adaptive max-pool out
// of LDS with fully coalesced non-temporal global stores.
__global__ __launch_bounds__(TPB) void roi_pool_kernel(
    const float* __restrict__ features,  // [1, 256, 7, 7]
    const int*   __restrict__ rois,      // [n_rois, 4] (x1,y1,x2,y2) * 32
    float*       __restrict__ out)       // [n_rois, 256, 7, 7]
{
    __shared__ float    sfeat[FEAT + 16];   // +16: speculative 2x2 over-read pad
    __shared__ unsigned sbin[OUTP * OUTP];

    const unsigned tid = threadIdx.x;
    const unsigned bid = blockIdx.x;

    // ---- async DMA of the feature map into LDS ----
    // 12 uniform b128 transfers per thread (12288 floats) + 1 b32 tail (256).
    // INST_OFFSET applies to both LDS and global addresses, addresses per-lane.
    const unsigned lds_base = (unsigned)(size_t)(void*)sfeat;
    {
        const float* gsrc = features;
        #pragma unroll
        for (unsigned j = 0; j < 12; ++j) {
            unsigned i        = tid + j * 256u;
            unsigned lds_addr = lds_base + i * 16u;
            unsigned goff     = i * 16u;
            asm volatile("global_load_async_to_lds_b128 %0, %1, %2"
                         :: "v"(lds_addr), "v"(goff), "s"(gsrc)
                         : "memory");
        }
        unsigned i        = 12288u + tid;        // remaining 256 floats
        unsigned lds_addr = lds_base + i * 4u;
        unsigned goff     = i * 4u;
        asm volatile("global_load_async_to_lds_b32 %0, %1, %2"
                     :: "v"(lds_addr), "v"(goff), "s"(gsrc)
                     : "memory");
    }

    // ---- adaptive-pool bin table (49 entries, uniform per block) ----
    // For L in [1,7], every bin is 1 or 2 wide in each axis and never empty:
    // pack start element offset (hs*7+ws, 6 bits) + has-2nd-col + has-2nd-row.
    const int x1 = rois[bid * 4 + 0] / SCALE;   // H-axis (torch dim 2)
    const int y1 = rois[bid * 4 + 1] / SCALE;   // W-axis (torch dim 3)
    const int x2 = rois[bid * 4 + 2] / SCALE;
    const int y2 = rois[bid * 4 + 3] / SCALE;
    const int Lh = x2 - x1;
    const int Lw = y2 - y1;

    if (tid < OUTP * OUTP) {
        const int oh = (int)tid / OUTP;
        const int ow = (int)tid % OUTP;
        const int hs = x1 + (oh * Lh) / OUTP;
        const int he = x1 + ((oh + 1) * Lh + OUTP - 1) / OUTP;
        const int ws = y1 + (ow * Lw) / OUTP;
        const int we = y1 + ((ow + 1) * Lw + OUTP - 1) / OUTP;
        unsigned pk = (unsigned)(hs * FM + ws);
        if (we - ws > 1) pk |= 256u;   // second column valid
        if (he - hs > 1) pk |= 512u;   // second row valid
        sbin[tid] = pk;
    }

    // Fence this wave's async traffic, then make LDS visible block-wide.
    asm volatile("s_wait_asynccnt 0x0" ::: "memory");
    __syncthreads();

    // ---- pooled output: exactly 49 iterations (12544 = 49 * 256) ----
    // Element idx = tid + 256*j -> channel c, bin r advance incrementally:
    // 256 = 5*49 + 11, so r += 11, c += 5 (+carry).
    unsigned c    = tid / 49u;           // one-time division
    unsigned r    = tid - c * 49u;
    unsigned coff = c * 49u;             // element offset of channel base
    float* dptr = out + (size_t)bid * OUT_PER_ROI + tid;

    #pragma unroll 7
    for (unsigned j = 0; j < 49u; ++j) {
        const unsigned b  = sbin[r];
        const float*   p  = sfeat + coff + (b & 63u);

        // Branchless 2x2 window max (corners outside the bin replaced by p[0]).
        const float a0 = p[0];
        const float a1 = p[1];
        const float a2 = p[FM];
        const float a3 = p[FM + 1];
        const bool  w2 = (b & 256u) != 0u;
        const bool  h2 = (b & 512u) != 0u;
        float m = a0;
        m = fmaxf(m, w2 ? a1 : a0);
        m = fmaxf(m, h2 ? a2 : a0);
        m = fmaxf(m, (w2 && h2) ? a3 : a0);

        __builtin_nontemporal_store(m, dptr + j * 256u);  // coalesced, streaming

        // advance (c, r) for next idx += 256
        r += 11u;
        coff += 245u;
        if (r >= 49u) { r -= 49u; coff += 49u; }
    }
}

extern "C" void kernel_launch(void* const* d_in, const int* in_sizes, int n_in,
                              void* d_out, int out_size, void* d_ws, size_t ws_size,
                              hipStream_t stream) {
    const float* features = (const float*)d_in[0];
    const int*   rois     = (const int*)d_in[1];
    float*       out      = (float*)d_out;

    const int n_rois = in_sizes[1] / 4;  // 4*512 = 2048 boxes
    roi_pool_kernel<<<n_rois, TPB, 0, stream>>>(features, rois, out);
}